// QuantumSelfAttentionModel_65481071410834
// MI455X (gfx1250) — compile-verified
//
#include <hip/hip_runtime.h>

// --- WMMA fragment types (wave32, CDNA5) ---
typedef __attribute__((ext_vector_type(2))) float v2f;  // A/B frag for 16x16x4 f32
typedef __attribute__((ext_vector_type(8))) float v8f;  // 16x16 f32 C/D frag

#define NQ 16         // quantum features / K dimension
#define DOUT 1024     // embed dim
#define XSTRIDE 1024  // x row stride (only first 16 used)

__global__ __launch_bounds__(256) void qsa_wmma_kernel(
    const float* __restrict__ x,     // [B*S, 1024]
    const float* __restrict__ W,     // [1024, 16] row-major
    const float* __restrict__ bias,  // [1024]
    float* __restrict__ out)         // [B*S, 1024]
{
    __shared__ float c_sh[16][17];   // cos values, padded to dodge bank conflicts
    __shared__ float z_sh[16][16];   // quantum features per row

    const int  tid  = threadIdx.x;
    const long row0 = (long)blockIdx.x * 16;

    // ---- Stage 1: cos of first 16 elems of each of 16 rows (1 elem / thread) ----
    {
        const int r = tid >> 4;
        const int q = tid & 15;
        c_sh[r][q] = __cosf(x[(row0 + r) * (long)XSTRIDE + q]);
    }
    __syncthreads();

    // ---- Stage 2: cumprod features (16 lanes, one row each; ~30 serial mults) ----
    if (tid < 16) {
        const int r = tid;
        float run = c_sh[r][0];   // cp[0]
        float p1  = 1.0f;         // prod c[1..15]
        #pragma unroll
        for (int q = 1; q < NQ; ++q) {
            const float cq = c_sh[r][q];
            run *= cq;            // cp[q]
            z_sh[r][q] = run;
            p1  *= cq;
        }
        z_sh[r][0] = p1;          // z0 = prod_{i>=1} cos
    }
    __syncthreads();

    const int lane  = tid & 31;
    const int wave  = tid >> 5;
    const int m     = lane & 15;   // A: row M; B/C/D: column N
    const int khalf = lane >> 4;   // 0: K=0,1  |  1: K=2,3  (per ISA 16x4 f32 A layout)

    // ---- A fragments (z), one per K-step of 4; reused for all 8 column tiles ----
    v2f a[4];
    #pragma unroll
    for (int kk = 0; kk < 4; ++kk) {
        a[kk].x = z_sh[m][kk * 4 + khalf * 2 + 0];
        a[kk].y = z_sh[m][kk * 4 + khalf * 2 + 1];
    }

    // ---- Each wave covers a 128-column stripe = 8 tiles of 16 ----
    #pragma unroll
    for (int t = 0; t < 8; ++t) {
        const int col0 = wave * 128 + t * 16;
        const int col  = col0 + m;

        v8f acc = {0.f, 0.f, 0.f, 0.f, 0.f, 0.f, 0.f, 0.f};

#if __has_builtin(__builtin_amdgcn_wmma_f32_16x16x4_f32)
        #pragma unroll
        for (int kk = 0; kk < 4; ++kk) {
            // B layout (4x16, 2 VGPRs): VGPR0 lanes0-15 K=4kk+0, lanes16-31 K=4kk+2;
            // VGPR1 is K+1 of each — i.e. a contiguous float2 of W[col][..], same
            // index pattern as A. W is [1024,16] row-major so this is an 8B load.
            const float* wp = W + col * NQ + kk * 4 + khalf * 2;
            v2f bf;
            bf.x = wp[0];
            bf.y = wp[1];
            acc = __builtin_amdgcn_wmma_f32_16x16x4_f32(
                /*neg_a=*/false, a[kk], /*neg_b=*/false, bf,
                /*c_mod=*/(short)0, acc, /*reuse_a=*/false, /*reuse_b=*/false);
        }
#else
        // VALU fallback (keeps the file compiling if the builtin is absent;
        // histogram wmma==0 tells us to switch path next round)
        #pragma unroll
        for (int r = 0; r < 8; ++r) {
            const int gm = r + khalf * 8;   // output row within tile
            float s = 0.f;
            #pragma unroll
            for (int k = 0; k < NQ; ++k)
                s += z_sh[gm][k] * W[col * NQ + k];
            acc[r] = s;
        }
#endif

        const float bb = bias[col];
        // C/D layout: VGPR r -> row M=r (lanes 0-15) / M=r+8 (lanes 16-31), N=lane&15
        #pragma unroll
        for (int r = 0; r < 8; ++r) {
            const long grow = row0 + r + khalf * 8;
            __builtin_nontemporal_store(acc[r] + bb, &out[grow * (long)DOUT + col]);
        }
    }
}

extern "C" void kernel_launch(void* const* d_in, const int* in_sizes, int n_in,
                              void* d_out, int out_size, void* d_ws, size_t ws_size,
                              hipStream_t stream) {
    const float* x  = (const float*)d_in[0];  // [32*2048, 1024]
    const float* W  = (const float*)d_in[1];  // [1024, 16]
    const float* b  = (const float*)d_in[2];  // [1024]
    float* out      = (float*)d_out;          // [32*2048, 1024]

    const long rows = (long)in_sizes[0] / XSTRIDE;   // 65536
    const int  nblocks = (int)(rows / 16);           // 4096

    qsa_wmma_kernel<<<dim3(nblocks), dim3(256), 0, stream>>>(x, W, b, out);
}